// RNN_50989851738801
// MI455X (gfx1250) — compile-verified
//
#include <hip/hip_runtime.h>
#include <hip/hip_bf16.h>

// Problem sizes (match reference)
#define SEQ   65536
#define DIN   256
#define HID   128

typedef float v2f __attribute__((ext_vector_type(2)));
typedef float v8f __attribute__((ext_vector_type(8)));
typedef int   gv4i __attribute__((vector_size(16)));   // matches builtin param type

#define AS1 __attribute__((address_space(1)))
#define AS3 __attribute__((address_space(3)))

// Feature probes (compile-safe: fall back to plain load/store staging)
#if __has_builtin(__builtin_amdgcn_global_load_async_to_lds_b128)
#define HAVE_ASYNC_LDS 1
#endif
#if __has_builtin(__builtin_amdgcn_s_wait_asynccnt)
#define HAVE_WAIT_ASYNC 1
#endif

// generic -> AS-qualified typed pointers via integer round-trip
__device__ __forceinline__ AS1 gv4i* glb_v4i(const void* p) {
    return (AS1 gv4i*)(__UINTPTR_TYPE__)p;               // 64-bit global pointer
}
__device__ __forceinline__ AS3 gv4i* lds_v4i(void* p) {
    return (AS3 gv4i*)(unsigned int)(__UINTPTR_TYPE__)p; // 32-bit LDS offset
}

__device__ __forceinline__ float fast_tanh(float x) {
#if __has_builtin(__builtin_amdgcn_tanhf)
    return __builtin_amdgcn_tanhf(x);
#elif __has_builtin(__builtin_amdgcn_tanh_f32)
    return __builtin_amdgcn_tanh_f32(x);
#else
    return tanhf(x);
#endif
}

// ---------------------------------------------------------------------------
// Kernel 1: xW = input @ Wi^T + bi   (fp32 WMMA 16x16x4, memory-bound GEMM)
//
// Per block: 256 threads = 8 waves, BM=32 rows of input staged in LDS via
// GLOBAL_LOAD_ASYNC_TO_LDS_B128 (ASYNCcnt path) when available.
// Wave w computes output columns [16w, 16w+16) for two 16-row M-subtiles.
// WMMA fragment layouts (wave32, 32-bit data):
//   A 16x4 : lane L -> row m = L%16, VGPR v holds K = (L/16)*2 + v
//   B 4x16 : lane L -> col n = L%16, VGPR v holds K = (L/16)*2 + v
//   C 16x16: lane L -> col n = L%16, VGPR v holds row m = v + 8*(L/16)
// ---------------------------------------------------------------------------
__global__ __launch_bounds__(256) void rnn_xw_wmma(
    const float* __restrict__ x,    // [SEQ, DIN]
    const float* __restrict__ Wi,   // [HID, DIN]
    const float* __restrict__ bi,   // [HID]
    float* __restrict__ out)        // [SEQ, HID] <- xW + bi
{
    constexpr int BM = 32;
    constexpr int LS = DIN + 12;     // 268 dwords: 16B-aligned rows, conflict-free frags
    __shared__ float As[BM * LS];

    const int t  = threadIdx.x;
    const long s0 = (long)blockIdx.x * BM;

    // ---- stage A tile (32 x 256 f32 = 32KB) into LDS ----
    {
        const float4* __restrict__ src = (const float4*)(x + s0 * DIN);
        #pragma unroll
        for (int i = 0; i < (BM * DIN / 4) / 256; ++i) {   // 8 iters
            int idx = t + i * 256;          // 0..2047
            int row = idx >> 6;             // / (DIN/4)
            int c4  = idx & 63;
#if HAVE_ASYNC_LDS
            // memory -> LDS without touching VGPRs; tracked by ASYNCcnt
            __builtin_amdgcn_global_load_async_to_lds_b128(
                glb_v4i(src + idx), lds_v4i(&As[row * LS + c4 * 4]),
                /*imm offset*/0, /*cpol*/0);
#else
            float4 v = src[idx];
            *(float4*)&As[row * LS + c4 * 4] = v;
#endif
        }
    }
#if HAVE_ASYNC_LDS
#if HAVE_WAIT_ASYNC
    __builtin_amdgcn_s_wait_asynccnt(0);
#else
    asm volatile("s_wait_asynccnt 0x0" ::: "memory");
#endif
#endif
    __syncthreads();

    const int lane = t & 31;
    const int wave = t >> 5;        // 0..7  -> n0 = 16*wave
    const int half = lane >> 4;     // 0,1
    const int r    = lane & 15;
    const int n0   = wave * 16;

    // accumulators pre-loaded with bias (D[m][n] column n = n0+r for all 8 regs)
    const float bias = bi[n0 + r];
    v8f c0, c1;
    #pragma unroll
    for (int v = 0; v < 8; ++v) { c0[v] = bias; c1[v] = bias; }

    // B[k][n] = Wi[n0+n][kb+k]; lane holds contiguous pair -> float2 loads
    const float* __restrict__ wrow  = Wi + (size_t)(n0 + r) * DIN + half * 2;
    const float* __restrict__ arow0 = &As[(r      ) * LS + half * 2];
    const float* __restrict__ arow1 = &As[(16 + r ) * LS + half * 2];

    #pragma unroll 4
    for (int kb = 0; kb < DIN / 4; ++kb) {      // 64 K-steps of 4
        v2f b  = *(const v2f*)(wrow  + kb * 4);
        v2f a0 = *(const v2f*)(arow0 + kb * 4);
        v2f a1 = *(const v2f*)(arow1 + kb * 4);
        c0 = __builtin_amdgcn_wmma_f32_16x16x4_f32(false, a0, false, b,
                                                   (short)0, c0, false, false);
        c1 = __builtin_amdgcn_wmma_f32_16x16x4_f32(false, a1, false, b,
                                                   (short)0, c1, false, false);
    }

    // ---- store D tiles: out[(s0+m)*HID + n0 + r], m = v + 8*half (+16 for c1)
    {
        float* __restrict__ o0 = out + (s0 + 8 * half) * HID + n0 + r;
        #pragma unroll
        for (int v = 0; v < 8; ++v) o0[(size_t)v * HID] = c0[v];
        float* __restrict__ o1 = out + (s0 + 16 + 8 * half) * HID + n0 + r;
        #pragma unroll
        for (int v = 0; v < 8; ++v) o1[(size_t)v * HID] = c1[v];
    }
}

// ---------------------------------------------------------------------------
// Kernel 2: sequential scan  h_t = tanh(xW_t + h_{t-1} @ Wh^T + bh)
//
// One block, 128 threads = 4 wave32 (one per SIMD of a WGP).
// Thread j owns Wh row j in 128 VGPRs; h kept in LDS, read via uniform-address
// (hardware broadcast) ds_load_b128; ping-pong h buffers -> 1 barrier/step.
// xW is read from `out` (written by kernel 1) in batches of 8 timesteps and
// overwritten in place with h_t. Prefetch runs one batch ahead.
// ---------------------------------------------------------------------------
__global__ __launch_bounds__(128) void rnn_scan(
    const float* __restrict__ h0,   // [HID]
    const float* __restrict__ Wh,   // [HID, HID]
    const float* __restrict__ bh,   // [HID]
    float* __restrict__ out,        // [SEQ, HID] in: xW+bi, out: h_t
    float* __restrict__ hn)         // [HID] final hidden
{
    const int j = threadIdx.x;          // 0..127
    __shared__ float hbuf[2][HID];

    // Wh row j -> registers (128 VGPRs)
    float w[HID];
    #pragma unroll
    for (int k = 0; k < HID; ++k) w[k] = Wh[(size_t)j * HID + k];
    const float bhj = bh[j];

    hbuf[0][j] = h0[j];
    __syncthreads();

    int p = 0;
    float hlast = 0.0f;

    for (int sb = 0; sb < SEQ; sb += 8) {
        // batch-load 8 timesteps of xW (independent loads, one wait)
        float xw[8];
        #pragma unroll
        for (int i = 0; i < 8; ++i)
            xw[i] = out[(size_t)(sb + i) * HID + j];

        // prefetch next batch into caches (global_prefetch_b8)
        if (sb + 8 < SEQ)
            __builtin_prefetch(&out[(size_t)(sb + 8) * HID + j], 0, 0);

        #pragma unroll
        for (int i = 0; i < 8; ++i) {
            float acc = xw[i] + bhj;
            const float4* __restrict__ hv = (const float4*)hbuf[p];
            #pragma unroll
            for (int k4 = 0; k4 < HID / 4; ++k4) {      // 32 x ds_load_b128 (broadcast)
                float4 h4 = hv[k4];
                acc = fmaf(w[4 * k4 + 0], h4.x, acc);
                acc = fmaf(w[4 * k4 + 1], h4.y, acc);
                acc = fmaf(w[4 * k4 + 2], h4.z, acc);
                acc = fmaf(w[4 * k4 + 3], h4.w, acc);
            }
            hlast = fast_tanh(acc);
            out[(size_t)(sb + i) * HID + j] = hlast;    // overwrite xW with h_t
            hbuf[p ^ 1][j] = hlast;                     // write NEW buffer
            p ^= 1;
            __syncthreads();                            // 1 barrier per step
        }
    }

    hn[j] = hlast;
}

// ---------------------------------------------------------------------------
// Launch: inputs are (input, h_0, Wi, bi, Wh, bh); d_out = [output S*H | h_n H]
// ---------------------------------------------------------------------------
extern "C" void kernel_launch(void* const* d_in, const int* in_sizes, int n_in,
                              void* d_out, int out_size, void* d_ws, size_t ws_size,
                              hipStream_t stream) {
    const float* x  = (const float*)d_in[0];
    const float* h0 = (const float*)d_in[1];
    const float* Wi = (const float*)d_in[2];
    const float* bi = (const float*)d_in[3];
    const float* Wh = (const float*)d_in[4];
    const float* bh = (const float*)d_in[5];
    float* out = (float*)d_out;

    rnn_xw_wmma<<<SEQ / 32, 256, 0, stream>>>(x, Wi, bi, out);
    rnn_scan<<<1, 128, 0, stream>>>(h0, Wh, bh, out, out + (size_t)SEQ * HID);
}